// ChamferDistanceWithPolarity_63127429316930
// MI455X (gfx1250) — compile-verified
//
#include <hip/hip_runtime.h>
#include <math.h>

typedef __attribute__((ext_vector_type(2))) float v2f;
typedef __attribute__((ext_vector_type(8))) float v8f;

#define N_PTS   16384
#define N_TILES (N_PTS / 16)        // 1024
#define SQRT_BETA 0.70710678118654752440f   // sqrt(0.5)

// Workspace layout in floats:
//   Q  : [0,        65536)   lifted pred  (16384 x 4)
//   R  : [65536,   131072)   lifted target(16384 x 4)
//   QN : [131072,  147456)   |q_i|^2
//   RN : [147456,  163840)   |r_j|^2
//   MP : [163840,  180224)   min over j of combined(i,j)   (p2g)
//   MG : [180224,  196608)   min over i of combined(i,j)   (g2p)
#define WS_Q   0
#define WS_R   65536
#define WS_QN  131072
#define WS_RN  147456
#define WS_MP  163840
#define WS_MG  180224

// ---------------------------------------------------------------------------
// 1) Lift points into 4-D (absorb BETA via sqrt into the polarity channel)
//    and precompute squared norms.
// ---------------------------------------------------------------------------
__global__ void __launch_bounds__(256)
chamfer_prep(const float* __restrict__ pred,
             const float* __restrict__ target,
             float* __restrict__ ws) {
    int i = blockIdx.x * blockDim.x + threadIdx.x;
    if (i >= N_PTS) return;

    float4 p = ((const float4*)pred)[i];
    float4 t = ((const float4*)target)[i];

    float4 q = make_float4(p.x, p.y, p.z, SQRT_BETA * p.w);
    float4 r = make_float4(t.x, t.y, t.z, SQRT_BETA * t.w);

    ((float4*)(ws + WS_Q))[i] = q;
    ((float4*)(ws + WS_R))[i] = r;
    ws[WS_QN + i] = q.x * q.x + q.y * q.y + q.z * q.z + q.w * q.w;
    ws[WS_RN + i] = r.x * r.x + r.y * r.y + r.z * r.z + r.w * r.w;
}

// ---------------------------------------------------------------------------
// 2) One wave per 16-row tile. Sweep all 1024 column tiles:
//      cross = Q_tile (16x4)  x  R_tile^T (4x16)   via V_WMMA_F32_16X16X4_F32
//      partial(m,n) = RN[n] - 2*cross(m,n) ;  running elementwise min
//    Then cross-lane min over the 16 columns, add QN[m], store per-row min.
//    blockIdx.y selects direction (0: pred->target, 1: target->pred).
// ---------------------------------------------------------------------------
__global__ void __launch_bounds__(256)
chamfer_min_pass(float* __restrict__ ws) {
    const int dir  = blockIdx.y;
    const float* __restrict__ Q  = ws + (dir ? WS_R  : WS_Q);
    const float* __restrict__ R  = ws + (dir ? WS_Q  : WS_R);
    const float* __restrict__ QN = ws + (dir ? WS_RN : WS_QN);
    const float* __restrict__ RN = ws + (dir ? WS_QN : WS_RN);
    float*       __restrict__ OUT = ws + (dir ? WS_MG : WS_MP);

    const int lane    = threadIdx.x & 31;
    const int wave    = (blockIdx.x * blockDim.x + threadIdx.x) >> 5;
    const int rowTile = wave;                      // 0 .. 1023
    const int lane15  = lane & 15;
    const int kb      = (lane < 16) ? 0 : 2;       // ISA A/B fragment layout

    // A fragment: row m = rowTile*16 + lane15, K = kb, kb+1  (2 VGPRs, loaded once)
    const int m = rowTile * 16 + lane15;
    v2f a = *(const v2f*)(Q + m * 4 + kb);

    v8f mmin;
#pragma unroll
    for (int v = 0; v < 8; ++v) mmin[v] = 3.402823466e38f;

#pragma unroll 4
    for (int t = 0; t < N_TILES; ++t) {
        const int n = t * 16 + lane15;
        // B fragment: col n, K = kb, kb+1 (8-byte aligned -> global_load_b64)
        v2f b = *(const v2f*)(R + n * 4 + kb);
        float rn = RN[n];

        v8f c = {};
        c = __builtin_amdgcn_wmma_f32_16x16x4_f32(
                /*neg_a=*/false, a, /*neg_b=*/false, b,
                /*c_mod=*/(short)0, c, /*reuse_a=*/false, /*reuse_b=*/false);

        // partial = RN[n] - 2*cross ; running min (QN[m] added after reduction)
#pragma unroll
        for (int v = 0; v < 8; ++v)
            mmin[v] = fminf(mmin[v], fmaf(c[v], -2.0f, rn));
    }

    // Min across the 16 lanes of each half-wave (C layout: rows 0-7 in lanes
    // 0-15, rows 8-15 in lanes 16-31; xor masks 1..8 stay within each half).
#pragma unroll
    for (int off = 1; off < 16; off <<= 1) {
#pragma unroll
        for (int v = 0; v < 8; ++v)
            mmin[v] = fminf(mmin[v], __shfl_xor(mmin[v], off, 32));
    }

    if (lane15 == 0) {
        const int rbase = rowTile * 16 + ((lane >> 4) << 3);  // lane0: rows 0-7, lane16: rows 8-15
#pragma unroll
        for (int v = 0; v < 8; ++v)
            OUT[rbase + v] = mmin[v] + QN[rbase + v];
    }
}

// ---------------------------------------------------------------------------
// 3) Final reduction: mean(MP) + mean(MG) = (sum MP + sum MG) / N
// ---------------------------------------------------------------------------
__global__ void __launch_bounds__(256)
chamfer_reduce(const float* __restrict__ ws, float* __restrict__ out) {
    const float* MP = ws + WS_MP;
    const float* MG = ws + WS_MG;

    float acc = 0.0f;
    for (int i = threadIdx.x; i < N_PTS; i += 256)
        acc += MP[i] + MG[i];

#pragma unroll
    for (int off = 16; off > 0; off >>= 1)
        acc += __shfl_xor(acc, off, 32);

    __shared__ float s[8];
    const int lane = threadIdx.x & 31;
    const int w    = threadIdx.x >> 5;
    if (lane == 0) s[w] = acc;
    __syncthreads();
    if (threadIdx.x == 0) {
        float t = 0.0f;
#pragma unroll
        for (int i = 0; i < 8; ++i) t += s[i];
        out[0] = t / (float)N_PTS;   // LOSS_WEIGHT = 1
    }
}

// ---------------------------------------------------------------------------
extern "C" void kernel_launch(void* const* d_in, const int* in_sizes, int n_in,
                              void* d_out, int out_size, void* d_ws, size_t ws_size,
                              hipStream_t stream) {
    const float* pred   = (const float*)d_in[0];
    const float* target = (const float*)d_in[1];
    float*       ws     = (float*)d_ws;
    float*       out    = (float*)d_out;

    // 1) lift + norms
    chamfer_prep<<<dim3((N_PTS + 255) / 256), dim3(256), 0, stream>>>(pred, target, ws);

    // 2) both min passes: 1024 waves per direction, 8 waves (256 thr) per block
    dim3 grid(N_TILES * 32 / 256, 2);
    chamfer_min_pass<<<grid, dim3(256), 0, stream>>>(ws);

    // 3) scalar result
    chamfer_reduce<<<dim3(1), dim3(256), 0, stream>>>(ws, out);
}